// Decoder_Layer_F_41188736368688
// MI455X (gfx1250) — compile-verified
//
#include <hip/hip_runtime.h>
#include <hip/hip_bf16.h>

// ---------------------------------------------------------------------------
// Decoder layer: unfold -> grouped ConvTranspose2d -> BatchNorm -> LeakyReLU -> fold
// B=64, NZ=128, NC=64, G=256 (16x16 blocks), latent patch 4x4 -> out patch 8x8
// Pipeline:
//   k1: per-(g, b-tile16, nc-tile16) WMMA f32 16x16x4 convT -> Y[g][b][p][nc]
//   k2: per-g batch stats -> scale/shift per channel (g*64+nc)
//   k3: per-(g,b) normalize + leaky + fold -> out[b][nc][128][128]
// ---------------------------------------------------------------------------

typedef float v2f __attribute__((ext_vector_type(2)));
typedef float v8f __attribute__((ext_vector_type(8)));

#define NZ_   128
#define NC_   64
#define G_    256
#define B_    64
#define XSTR  132                      // padded K stride (floats) -> conflict-free ds_load_b64
#define LDS_TILE_FLOATS (16 * 16 * XSTR)   // 33792 floats = 132 KB per operand tile

// ------------------------------- kernel 1 ----------------------------------
// grid = 256 g * 4 b-tiles * 4 nc-tiles = 4096 blocks, 256 threads (8 waves)
__global__ __launch_bounds__(256) void k1_convt_wmma(
    const float* __restrict__ X,   // [B][NZ][64][64]
    const float* __restrict__ W,   // [G*NZ][NC][4][4]
    float* __restrict__ Y)         // [G][B][64p][64nc]
{
    extern __shared__ float smem[];
    float* Xl = smem;                    // [p(16)][b(16)][nz : XSTR]
    float* Wl = smem + LDS_TILE_FLOATS;  // [t(16)][nc(16)][nz : XSTR]

    const int blk = blockIdx.x;
    const int g   = blk >> 4;
    const int bt  = (blk >> 2) & 3;
    const int nct = blk & 3;
    const int b0  = bt * 16;
    const int nc0 = nct * 16;
    const int gh  = g >> 4;
    const int gw  = g & 15;

    const int tid = threadIdx.x;

    // ---- stage X tile: 16 b x 128 nz x 16 positions (float4 over iw) ----
    // global row for (b,nz,ih): X[(b0+b)*128+nz][gh*4+ih][gw*4 .. +3]
    for (int i = tid; i < 16 * NZ_ * 4; i += 256) {
        const int b  = i >> 9;          // /512
        const int nz = (i >> 2) & 127;
        const int ih = i & 3;
        const float4 v = *(const float4*)(X + ((size_t)(b0 + b) * NZ_ + nz) * 4096
                                            + (gh * 4 + ih) * 64 + gw * 4);
        float* dst0 = Xl + (size_t)(ih * 4) * 16 * XSTR + b * XSTR + nz;
        dst0[0 * 16 * XSTR] = v.x;
        dst0[1 * 16 * XSTR] = v.y;
        dst0[2 * 16 * XSTR] = v.z;
        dst0[3 * 16 * XSTR] = v.w;
    }

    // ---- stage W tile: 128 nz x 16 nc x 16 taps (float4 over taps) ----
    for (int i = tid; i < NZ_ * 16 * 4; i += 256) {
        const int nz = i >> 6;
        const int nc = (i >> 2) & 15;
        const int t4 = i & 3;
        const float4 v = *(const float4*)(W + (((size_t)g * NZ_ + nz) * NC_ + nc0 + nc) * 16
                                            + t4 * 4);
        float* dst0 = Wl + (size_t)(t4 * 4) * 16 * XSTR + nc * XSTR + nz;
        dst0[0 * 16 * XSTR] = v.x;
        dst0[1 * 16 * XSTR] = v.y;
        dst0[2 * 16 * XSTR] = v.z;
        dst0[3 * 16 * XSTR] = v.w;
    }

    __syncthreads();

    // ---- WMMA compute: wave wv handles 8 output positions ----
    const int wv   = tid >> 5;
    const int lane = tid & 31;
    const int m    = lane & 15;
    const int hi   = lane >> 4;

    for (int pi = 0; pi < 8; ++pi) {
        const int p  = wv * 8 + pi;      // uniform per wave -> EXEC stays full
        const int oh = p >> 3;
        const int ow = p & 7;

        v8f c = {0.f, 0.f, 0.f, 0.f, 0.f, 0.f, 0.f, 0.f};

        for (int kh = 0; kh < 4; ++kh) {
            const int nh = oh + 1 - kh;
            if (nh & 1) continue;
            const int ih = nh >> 1;
            if (ih < 0 || ih >= 4) continue;
            for (int kw = 0; kw < 4; ++kw) {
                const int nw = ow + 1 - kw;
                if (nw & 1) continue;
                const int iw = nw >> 1;
                if (iw < 0 || iw >= 4) continue;

                const float* Ap = Xl + ((ih * 4 + iw) * 16 + m) * XSTR + 2 * hi;
                const float* Bp = Wl + ((kh * 4 + kw) * 16 + m) * XSTR + 2 * hi;

#pragma unroll 8
                for (int kz = 0; kz < NZ_; kz += 4) {
                    v2f a = *(const v2f*)(Ap + kz);   // ds_load_b64, conflict-free
                    v2f b = *(const v2f*)(Bp + kz);
                    c = __builtin_amdgcn_wmma_f32_16x16x4_f32(
                            /*neg_a=*/false, a, /*neg_b=*/false, b,
                            /*c_mod=*/(short)0, c,
                            /*reuse_a=*/false, /*reuse_b=*/false);
                }
            }
        }

        // D layout: vgpr r, lanes0-15 -> M=r, lanes16-31 -> M=r+8 ; N = lane&15
        // Y[((g*64 + b)*64 + p)*64 + nc], b = b0 + 8*hi + r, nc = nc0 + m
        float* yp = Y + (((size_t)g * B_ + b0 + 8 * hi) * 64 + p) * 64 + nc0 + m;
#pragma unroll
        for (int r = 0; r < 8; ++r)
            yp[(size_t)r * 64 * 64] = c[r];
    }
}

// ------------------------------- kernel 2 ----------------------------------
// grid = 256 (one per group), 256 threads. Stats over (b, p) per nc.
__global__ __launch_bounds__(256) void k2_stats(
    const float* __restrict__ Y,
    const float* __restrict__ gamma,
    const float* __restrict__ beta,
    float* __restrict__ scaleArr,
    float* __restrict__ shiftArr)
{
    __shared__ float redS[256];
    __shared__ float redQ[256];

    const int g   = blockIdx.x;
    const int tid = threadIdx.x;
    const int nc  = tid & 63;
    const int q   = tid >> 6;           // 4 batch chunks of 16

    const float* base = Y + ((size_t)g * B_ + q * 16) * 4096 + nc;
    float s = 0.f, s2 = 0.f;
    for (int b = 0; b < 16; ++b) {
        const float* row = base + (size_t)b * 4096;
        for (int p = 0; p < 64; ++p) {
            float v = row[p * 64];      // lanes span 64 consecutive nc -> coalesced
            s  += v;
            s2 += v * v;
        }
    }
    redS[tid] = s;
    redQ[tid] = s2;
    __syncthreads();

    if (tid < 64) {
        float S  = redS[tid] + redS[tid + 64] + redS[tid + 128] + redS[tid + 192];
        float Q  = redQ[tid] + redQ[tid + 64] + redQ[tid + 128] + redQ[tid + 192];
        const float inv_n = 1.0f / 4096.0f;
        float mean = S * inv_n;
        float var  = Q * inv_n - mean * mean;
        float rstd = rsqrtf(var + 1e-5f);
        float sc   = gamma[g * 64 + tid] * rstd;
        scaleArr[g * 64 + tid] = sc;
        shiftArr[g * 64 + tid] = beta[g * 64 + tid] - mean * sc;
    }
}

// ------------------------------- kernel 3 ----------------------------------
// grid = 256 g * 64 b = 16384 blocks, 256 threads. Normalize + leaky + fold.
__global__ __launch_bounds__(256) void k3_apply_fold(
    const float* __restrict__ Y,
    const float* __restrict__ scaleArr,
    const float* __restrict__ shiftArr,
    float* __restrict__ out)          // [B][NC][128][128]
{
    __shared__ float Yl[64 * 65];     // [p][nc] padded -> conflict-free transpose reads
    __shared__ float scl[64];
    __shared__ float shf[64];

    const int blk = blockIdx.x;
    const int g   = blk >> 6;
    const int b   = blk & 63;
    const int gh  = g >> 4;
    const int gw  = g & 15;
    const int tid = threadIdx.x;

    const float* ysrc = Y + ((size_t)g * B_ + b) * 4096;
    for (int i = tid; i < 4096; i += 256) {
        const int p  = i >> 6;
        const int nc = i & 63;
        Yl[p * 65 + nc] = ysrc[i];    // fully coalesced load
    }
    if (tid < 64) {
        scl[tid] = scaleArr[g * 64 + tid];
        shf[tid] = shiftArr[g * 64 + tid];
    }
    __syncthreads();

    for (int i = tid; i < 4096; i += 256) {
        const int nc = i >> 6;
        const int oh = (i >> 3) & 7;
        const int ow = i & 7;
        float z = Yl[(oh * 8 + ow) * 65 + nc] * scl[nc] + shf[nc];
        z = z > 0.f ? z : 0.2f * z;
        out[(((size_t)b * NC_ + nc) * 128 + gh * 8 + oh) * 128 + gw * 8 + ow] = z;
    }
}

// ------------------------------- launcher ----------------------------------
extern "C" void kernel_launch(void* const* d_in, const int* in_sizes, int n_in,
                              void* d_out, int out_size, void* d_ws, size_t ws_size,
                              hipStream_t stream) {
    const float* X     = (const float*)d_in[0];   // 64*128*64*64
    const float* W     = (const float*)d_in[1];   // 32768*64*4*4
    const float* gamma = (const float*)d_in[2];   // 16384
    const float* beta  = (const float*)d_in[3];   // 16384
    float* out = (float*)d_out;                   // 64*64*128*128

    float* Y        = (float*)d_ws;               // 256*64*64*64 floats = 256 MiB
    float* scaleArr = Y + (size_t)G_ * B_ * 64 * 64;
    float* shiftArr = scaleArr + G_ * NC_;

    const size_t lds1 = (size_t)2 * LDS_TILE_FLOATS * sizeof(float);  // 270336 B

    k1_convt_wmma<<<dim3(4096), dim3(256), lds1, stream>>>(X, W, Y);
    k2_stats<<<dim3(256), dim3(256), 0, stream>>>(Y, gamma, beta, scaleArr, shiftArr);
    k3_apply_fold<<<dim3(16384), dim3(256), 0, stream>>>(Y, scaleArr, shiftArr, out);
}